// SessionConv_35192962024015
// MI455X (gfx1250) — compile-verified
//
#include <hip/hip_runtime.h>
#include <hip/hip_bf16.h>

#define N_NODES 50000
#define N_EDGES 800000
#define EMB 100
#define EPSN 1e-12f

typedef __attribute__((ext_vector_type(2))) float v2f;
typedef __attribute__((ext_vector_type(8))) float v8f;

// ---------------- zero workspace ----------------
__global__ __launch_bounds__(256) void zero_f4_kernel(float4* __restrict__ p, long n4) {
  long i = (long)blockIdx.x * blockDim.x + threadIdx.x;
  if (i < n4) p[i] = make_float4(0.f, 0.f, 0.f, 0.f);
}

// ---------------- SpMM: wave32 per edge, lane = float4 chunk ----------------
__global__ __launch_bounds__(256) void spmm_kernel(
    const int* __restrict__ rows, const int* __restrict__ cols,
    const float* __restrict__ vals, const float* __restrict__ xin,
    float* __restrict__ xout) {
  int gid  = blockIdx.x * blockDim.x + threadIdx.x;
  int edge = gid >> 5;   // one wave32 per edge
  int lane = gid & 31;
  if (edge >= N_EDGES) return;
  int   r = rows[edge];
  int   c = cols[edge];
  float v = vals[edge];
  if (lane < EMB / 4) {  // 25 float4 chunks cover 100 floats (row stride 400B, 16B aligned)
    const float4 s = *(const float4*)(xin + (long)c * EMB + 4 * lane);
    float* dst = xout + (long)r * EMB + 4 * lane;
    unsafeAtomicAdd(dst + 0, v * s.x);   // native global_atomic_add_f32 in L2
    unsafeAtomicAdd(dst + 1, v * s.y);
    unsafeAtomicAdd(dst + 2, v * s.z);
    unsafeAtomicAdd(dst + 3, v * s.w);
  }
}

// ---------------- normalize rows per layer + a-weighted combine via WMMA ----------------
// One wave32 per node. K=4 layer reduction done with V_WMMA_F32_16X16X4_F32:
//   A[i][k] = x_k[node, base+i] * s_k   (16x4 tile; lanes 0-15 carry K=0,1, lanes 16-31 K=2,3)
//   B       = all-ones 4x16            (layout-independent since all entries equal)
//   D[i][j] = out[node, base+i]  for every column j.
// Extraction: column N=0 lives in lane 0 (M=0..7 in d0..d7) and lane 16 (M=8..15),
// each holding 8 contiguous outputs -> one predicate (fl==0) + two b128 stores.
__global__ __launch_bounds__(256) void norm_combine_kernel(
    const float* __restrict__ x0, const float* __restrict__ x1,
    const float* __restrict__ x2, const float* __restrict__ x3,
    const float* __restrict__ a, float* __restrict__ out) {
  const int lane = threadIdx.x & 31;
  const int node = blockIdx.x * (blockDim.x >> 5) + (threadIdx.x >> 5);
  if (node >= N_NODES) return;  // wave-uniform exit: EXEC stays all-ones for WMMA

  const float* xr[4];
  xr[0] = x0 + (long)node * EMB;
  xr[1] = x1 + (long)node * EMB;
  xr[2] = x2 + (long)node * EMB;
  xr[3] = x3 + (long)node * EMB;

  // per-layer scale s_l = a[l] / max(||row||_2, eps), via wave32 shuffle reduction
  float s[4];
#pragma unroll
  for (int l = 0; l < 4; ++l) {
    float acc = 0.f;
    for (int f = lane; f < EMB; f += 32) { float t = xr[l][f]; acc += t * t; }
#pragma unroll
    for (int off = 16; off > 0; off >>= 1) acc += __shfl_xor(acc, off, 32);
    s[l] = a[l] / fmaxf(sqrtf(acc), EPSN);
  }

  const bool hi = (lane >= 16);
  const int  fl = lane & 15;
  const float* pA = hi ? xr[2] : xr[0];
  const float* pB = hi ? xr[3] : xr[1];
  const float  sA = hi ? s[2] : s[0];
  const float  sB = hi ? s[3] : s[1];

  v2f bmat; bmat.x = 1.0f; bmat.y = 1.0f;     // all-ones B
  float* orow = out + (long)node * EMB;
  // lane 0 writes M=0..7 at orow+base, lane 16 writes M=8..15 at orow+base+8
  float* obase = orow + (hi ? 8 : 0);

#pragma unroll
  for (int base = 0; base < 96; base += 16) {
    v2f amat;
    amat.x = pA[base + fl] * sA;              // K = 0 (lo) / 2 (hi)
    amat.y = pB[base + fl] * sB;              // K = 1 (lo) / 3 (hi)
    v8f cacc = {};
    cacc = __builtin_amdgcn_wmma_f32_16x16x4_f32(
        /*neg_a=*/false, amat, /*neg_b=*/false, bmat,
        /*c_mod=*/(short)0, cacc, /*reuse_a=*/false, /*reuse_b=*/false);
    if (fl == 0) {  // lanes 0 and 16 together cover all 16 outputs of this group
      float4 lo4 = make_float4(cacc[0], cacc[1], cacc[2], cacc[3]);
      float4 hi4 = make_float4(cacc[4], cacc[5], cacc[6], cacc[7]);
      *(float4*)(obase + base)     = lo4;   // global_store_b128
      *(float4*)(obase + base + 4) = hi4;   // global_store_b128
    }
  }
  // VALU tail for features 96..99
  if (lane < EMB - 96) {
    int f = 96 + lane;
    orow[f] = xr[0][f] * s[0] + xr[1][f] * s[1] + xr[2][f] * s[2] + xr[3][f] * s[3];
  }
}

extern "C" void kernel_launch(void* const* d_in, const int* in_sizes, int n_in,
                              void* d_out, int out_size, void* d_ws, size_t ws_size,
                              hipStream_t stream) {
  const int*   adj_row = (const int*)d_in[0];
  const int*   adj_col = (const int*)d_in[1];
  const float* adj_val = (const float*)d_in[2];
  const float* emb     = (const float*)d_in[3];
  const float* a       = (const float*)d_in[4];
  float*       out     = (float*)d_out;

  const long rowElems = (long)N_NODES * EMB;     // 5,000,000 floats
  float* x1 = (float*)d_ws;
  float* x2 = x1 + rowElems;
  float* x3 = x2 + rowElems;

  // zero the three scatter-add targets (15M floats = 3.75M float4)
  long n4 = 3 * rowElems / 4;
  zero_f4_kernel<<<(unsigned)((n4 + 255) / 256), 256, 0, stream>>>((float4*)d_ws, n4);

  // 3 propagation layers; kernel boundaries give the required ordering
  unsigned spmmBlocks = (unsigned)(((long)N_EDGES * 32 + 255) / 256);  // 100,000 blocks
  spmm_kernel<<<spmmBlocks, 256, 0, stream>>>(adj_row, adj_col, adj_val, emb, x1);
  spmm_kernel<<<spmmBlocks, 256, 0, stream>>>(adj_row, adj_col, adj_val, x1,  x2);
  spmm_kernel<<<spmmBlocks, 256, 0, stream>>>(adj_row, adj_col, adj_val, x2,  x3);

  // normalize + weighted combine (wave per node, 8 waves/block)
  norm_combine_kernel<<<(N_NODES + 7) / 8, 256, 0, stream>>>(emb, x1, x2, x3, a, out);
}